// FlexConvolution_2568390443359
// MI455X (gfx1250) — compile-verified
//
#include <hip/hip_runtime.h>

#define B_    4
#define DIN   64
#define DP    3
#define DOUT  64
#define KNB   16
#define NPT   8192
#define NT    128       // n-columns per workgroup tile
#define GSTR  129       // padded LDS stride for G  (conflict-free column writes)
#define TSTR  65        // padded LDS stride for thetaA

typedef __attribute__((ext_vector_type(2))) float v2f;
typedef __attribute__((ext_vector_type(8))) float v8f;

__launch_bounds__(256)
__global__ void flexconv_wmma_kernel(const float* __restrict__ features,   // [B, DIN, N]
                                     const float* __restrict__ positions,  // [B, DP,  N]
                                     const int*   __restrict__ nbhd,       // [B, K,   N]
                                     const float* __restrict__ theta,      // [DP, DIN, DOUT]
                                     const float* __restrict__ pbias,      // [DIN, DOUT]
                                     const float* __restrict__ fbias,      // [DOUT]
                                     float*       __restrict__ out)        // [B, DOUT, N]
{
    extern __shared__ float smem[];
    float* G    = smem;                       // [256][GSTR]   (q*64+i , n_local)
    float* thA  = G   + 256 * GSTR;           // [256][TSTR]   augmented theta
    float* rel  = thA + 256 * TSTR;           // [3][KNB][NT]
    int*   idxs = (int*)(rel + 3 * KNB * NT); // [KNB][NT]

    const int t  = threadIdx.x;
    const int b  = blockIdx.x / (NPT / NT);
    const int n0 = (blockIdx.x % (NPT / NT)) * NT;

    // ---------------- Phase 1a: stage rel, indices, thetaA into LDS ----------------
    {
        const int nl = t & (NT - 1);
        const int kh = t >> 7;                // 0..1 : each half of threads does 8 k's
        const int ng = n0 + nl;
        const float pc0 = positions[((size_t)b * DP + 0) * NPT + ng];
        const float pc1 = positions[((size_t)b * DP + 1) * NPT + ng];
        const float pc2 = positions[((size_t)b * DP + 2) * NPT + ng];
#pragma unroll
        for (int kk = 0; kk < KNB / 2; ++kk) {
            const int k   = kh * (KNB / 2) + kk;
            const int idx = nbhd[((size_t)b * KNB + k) * NPT + ng];   // coalesced (lane<->n)
            idxs[k * NT + nl] = idx;
            rel[(0 * KNB + k) * NT + nl] = positions[((size_t)b * DP + 0) * NPT + idx] - pc0;
            rel[(1 * KNB + k) * NT + nl] = positions[((size_t)b * DP + 1) * NPT + idx] - pc1;
            rel[(2 * KNB + k) * NT + nl] = positions[((size_t)b * DP + 2) * NPT + idx] - pc2;
        }
        // Augmented theta: rows 0..191 = position_theta, rows 192..255 = position_bias
        for (int e = t; e < 256 * DOUT; e += 256) {
            const int r = e >> 6, c = e & 63;
            const float v = (r < DP * DIN) ? theta[e] : pbias[e - DP * DIN * DOUT];
            thA[r * TSTR + c] = v;
        }
    }
    __syncthreads();

    // ---------------- Phase 1b: G[(q,i), n] = sum_k relA[q,k,n] * f[i, idx[k,n]] ----
    {
        const int i = t & 63;
        const int s = t >> 6;                 // 0..3
        const float* frow = features + ((size_t)b * DIN + i) * NPT;
        for (int nl = s; nl < NT; nl += 4) {
            float a0 = 0.f, a1 = 0.f, a2 = 0.f, a3 = 0.f;
#pragma unroll
            for (int k = 0; k < KNB; ++k) {
                const int   idx = idxs[k * NT + nl];       // LDS broadcast within wave
                const float f   = frow[idx];               // gathered, L2-resident
                a0 += rel[(0 * KNB + k) * NT + nl] * f;
                a1 += rel[(1 * KNB + k) * NT + nl] * f;
                a2 += rel[(2 * KNB + k) * NT + nl] * f;
                a3 += f;                                   // relA[3] == 1 (bias row)
            }
            G[(0 * DIN + i) * GSTR + nl] = a0;
            G[(1 * DIN + i) * GSTR + nl] = a1;
            G[(2 * DIN + i) * GSTR + nl] = a2;
            G[(3 * DIN + i) * GSTR + nl] = a3;
        }
    }
    __syncthreads();

    // ---------------- Phase 2: Y = thA^T (64x256) x G (256x128) via WMMA f32 --------
    // k0 outer / n-tile inner: each A-fragment is loaded once from LDS and reused
    // by 4 WMMAs (4 accumulators live in VGPRs).
    {
        const int wave  = t >> 5;
        const int lane  = t & 31;
        const int m0    = (wave & 3) * 16;    // o-tile base
        const int ntsel = wave >> 2;          // 0/1 -> n-tiles {sel, sel+2, sel+4, sel+6}
        const int mrow  = lane & 15;          // M (A) / N (B,D) index within tile
        const int khalf = lane >> 4;          // K-pair selector per ISA A layout

        // Hoist feature bias: D layout VGPR r -> M = r + 8*(lane/16)
        float fb[8];
#pragma unroll
        for (int r = 0; r < 8; ++r) fb[r] = fbias[m0 + r + 8 * khalf];

        v8f acc[4];
#pragma unroll
        for (int j = 0; j < 4; ++j)
#pragma unroll
            for (int r = 0; r < 8; ++r) acc[j][r] = fb[r];

#pragma unroll 2
        for (int k0 = 0; k0 < 256; k0 += 4) {
            const int ka = k0 + 2 * khalf;
            v2f av;
            av.x = thA[(ka + 0) * TSTR + m0 + mrow];       // A[m, k]  (16x4), loaded once
            av.y = thA[(ka + 1) * TSTR + m0 + mrow];
#pragma unroll
            for (int j = 0; j < 4; ++j) {
                const int nb16 = (ntsel + 2 * j) * 16;
                v2f bv;
                bv.x = G[(ka + 0) * GSTR + nb16 + mrow];   // B[k, n]  (4x16)
                bv.y = G[(ka + 1) * GSTR + nb16 + mrow];
                acc[j] = __builtin_amdgcn_wmma_f32_16x16x4_f32(
                            /*neg_a=*/false, av, /*neg_b=*/false, bv,
                            /*c_mod=*/(short)0, acc[j],
                            /*reuse_a=*/false, /*reuse_b=*/false);
            }
        }

#pragma unroll
        for (int j = 0; j < 4; ++j) {
            const int nb16 = (ntsel + 2 * j) * 16;
#pragma unroll
            for (int r = 0; r < 8; ++r) {
                const int o = m0 + r + 8 * khalf;
                out[((size_t)b * DOUT + o) * NPT + n0 + nb16 + mrow] = acc[j][r];
            }
        }
    }
}

extern "C" void kernel_launch(void* const* d_in, const int* in_sizes, int n_in,
                              void* d_out, int out_size, void* d_ws, size_t ws_size,
                              hipStream_t stream) {
    const float* features  = (const float*)d_in[0];
    const float* positions = (const float*)d_in[1];
    const int*   nbhd      = (const int*)  d_in[2];
    const float* theta     = (const float*)d_in[3];
    const float* pbias     = (const float*)d_in[4];
    const float* fbias     = (const float*)d_in[5];
    float* out = (float*)d_out;

    const size_t lds_bytes =
        (size_t)(256 * GSTR + 256 * TSTR + 3 * KNB * NT + KNB * NT) * sizeof(float);

    dim3 grid(B_ * (NPT / NT));   // 4 batches x 64 n-tiles = 256 workgroups
    dim3 block(256);              // 8 waves (wave32)
    flexconv_wmma_kernel<<<grid, block, lds_bytes, stream>>>(
        features, positions, nbhd, theta, pbias, fbias, out);
    (void)in_sizes; (void)n_in; (void)out_size; (void)d_ws; (void)ws_size;
}